// iSogCLR_New_v1_Loss_9972914061429
// MI455X (gfx1250) — compile-verified
//
#include <hip/hip_runtime.h>
#include <math.h>

// ---------------------------------------------------------------------------
// iSogCLR loss on MI455X (gfx1250, wave32).
//   pass1: sim tile via V_WMMA_F32_16X16X4_F32 -> row/col maxes (atomic max)
//   finalize_b: b_I/b_T stabilizers
//   pass2: recompute sim tiles -> masked exp sums (Sum e, Sum e*sim) per row/col
//   diag + final: O(B) scalar finalization in fp64
// ---------------------------------------------------------------------------

typedef __attribute__((ext_vector_type(2))) float v2f;
typedef __attribute__((ext_vector_type(8))) float v8f;

#define KC    64          // K-chunk staged in LDS
#define LSTR  68          // LDS row stride (floats): 64 + 4 pad -> 16B aligned rows, no bank conflicts
#define RHO_C 6.0f
#define GAM   0.8f

__device__ __forceinline__ void atomicMaxF(float* a, float v) {
  __hip_atomic_fetch_max(a, v, __ATOMIC_RELAXED, __HIP_MEMORY_SCOPE_AGENT);
}

// Compute a 64x64 tile of sim = img @ txt^T with fp32 WMMA.
// Wave w owns rows [w*16, w*16+16) of the tile, 4 column sub-tiles -> acc[0..3].
__device__ __forceinline__ void sim_tile_64x64(
    const float* __restrict__ img, const float* __restrict__ txt,
    int D, int gRow0, int gCol0, float* sA, float* sB, v8f acc[4]) {
  const int tid  = threadIdx.x;         // 0..127
  const int lane = tid & 31;
  const int wave = tid >> 5;
  const int m     = lane & 15;          // M (for A) / N (for B) within sub-tile
  const int kksel = (lane >> 4) << 1;   // upper half-wave carries K+2,K+3

  #pragma unroll
  for (int c = 0; c < 4; ++c)
    #pragma unroll
    for (int r = 0; r < 8; ++r) acc[c][r] = 0.0f;

  for (int kc = 0; kc < D; kc += KC) {
    // Stage 64 rows x 64 cols of each feature matrix into LDS (float4 strips).
    #pragma unroll
    for (int it = 0; it < 8; ++it) {
      int idx = it * 128 + tid;         // 0..1023 float4 slots
      int row = idx >> 4;
      int c4  = (idx & 15) << 2;
      *(float4*)(sA + row * LSTR + c4) =
          *(const float4*)(img + (size_t)(gRow0 + row) * D + kc + c4);
      *(float4*)(sB + row * LSTR + c4) =
          *(const float4*)(txt + (size_t)(gCol0 + row) * D + kc + c4);
    }
    __syncthreads();

    const float* aBase = sA + (wave * 16 + m) * LSTR;
    #pragma unroll
    for (int k = 0; k < KC; k += 4) {
      const int kk = k + kksel;
      const v2f a = *(const v2f*)(aBase + kk);          // A: 16x4 fragment
      #pragma unroll
      for (int c = 0; c < 4; ++c) {
        const v2f b = *(const v2f*)(sB + (c * 16 + m) * LSTR + kk);  // B: 4x16 fragment
        acc[c] = __builtin_amdgcn_wmma_f32_16x16x4_f32(
            false, a, false, b, (short)0, acc[c], false, false);
      }
    }
    __syncthreads();
  }
}

__global__ void init_ws_kernel(float* rowmax, float* colmax, float* GI, float* AI,
                               float* GT, float* AT, int B) {
  int i = blockIdx.x * blockDim.x + threadIdx.x;
  if (i < B) {
    rowmax[i] = -__builtin_inff();
    colmax[i] = -__builtin_inff();
    GI[i] = 0.0f; AI[i] = 0.0f; GT[i] = 0.0f; AT[i] = 0.0f;
  }
}

__global__ void __launch_bounds__(128)
pass1_maxes(const float* __restrict__ img, const float* __restrict__ txt, int D,
            float* __restrict__ rowmax, float* __restrict__ colmax) {
  __shared__ float sA[64 * LSTR];
  __shared__ float sB[64 * LSTR];
  const int gRow0 = blockIdx.x * 64;
  const int gCol0 = blockIdx.y * 64;
  v8f acc[4];
  sim_tile_64x64(img, txt, D, gRow0, gCol0, sA, sB, acc);

  const int lane = threadIdx.x & 31;
  const int wave = threadIdx.x >> 5;

  // Row maxes: reduce across the 16 N-lanes of each half-wave.
  #pragma unroll
  for (int r = 0; r < 8; ++r) {
    float v = fmaxf(fmaxf(acc[0][r], acc[1][r]), fmaxf(acc[2][r], acc[3][r]));
    v = fmaxf(v, __shfl_xor(v, 8, 32));
    v = fmaxf(v, __shfl_xor(v, 4, 32));
    v = fmaxf(v, __shfl_xor(v, 2, 32));
    v = fmaxf(v, __shfl_xor(v, 1, 32));
    if ((lane & 15) == 0)
      atomicMaxF(rowmax + gRow0 + wave * 16 + r + ((lane >> 4) << 3), v);
  }
  // Col maxes: fold the two M-halves, then one atomic per column.
  #pragma unroll
  for (int c = 0; c < 4; ++c) {
    float v = acc[c][0];
    #pragma unroll
    for (int r = 1; r < 8; ++r) v = fmaxf(v, acc[c][r]);
    v = fmaxf(v, __shfl_xor(v, 16, 32));
    if (lane < 16) atomicMaxF(colmax + gCol0 + c * 16 + lane, v);
  }
}

__global__ void finalize_b_kernel(const float* rowmax, const float* colmax,
                                  const float* tauI, const float* tauT,
                                  const float* b_I, const float* b_T,
                                  const int* iid, const int* tid_,
                                  float* bIn, float* bTn,
                                  float* oBI, float* oBT, int B) {
  int i = blockIdx.x * blockDim.x + threadIdx.x;
  if (i < B) {
    float obi = b_I[iid[i]];
    float obt = b_T[tid_[i]];
    bIn[i] = fmaxf(rowmax[i] / tauI[i], obi);
    bTn[i] = fmaxf(colmax[i] / tauT[i], obt);
    oBI[i] = obi;
    oBT[i] = obt;
  }
}

__global__ void __launch_bounds__(128)
pass2_sums(const float* __restrict__ img, const float* __restrict__ txt, int D,
           const float* __restrict__ tauI, const float* __restrict__ tauT,
           const float* __restrict__ bIn, const float* __restrict__ bTn,
           float* __restrict__ GI, float* __restrict__ AI,
           float* __restrict__ GT, float* __restrict__ AT) {
  __shared__ float sA[64 * LSTR];
  __shared__ float sB[64 * LSTR];
  const int gRow0 = blockIdx.x * 64;
  const int gCol0 = blockIdx.y * 64;
  v8f acc[4];
  sim_tile_64x64(img, txt, D, gRow0, gCol0, sA, sB, acc);

  const int lane = threadIdx.x & 31;
  const int wave = threadIdx.x >> 5;
  const int rowBaseW = gRow0 + wave * 16 + ((lane >> 4) << 3);  // this lane's 8 rows
  const int colL = lane & 15;

  float itI[8], bI8[8];
  #pragma unroll
  for (int r = 0; r < 8; ++r) {
    itI[r] = 1.0f / tauI[rowBaseW + r];
    bI8[r] = bIn[rowBaseW + r];
  }

  float gIs[8], aIs[8];
  #pragma unroll
  for (int r = 0; r < 8; ++r) { gIs[r] = 0.0f; aIs[r] = 0.0f; }

  #pragma unroll
  for (int c = 0; c < 4; ++c) {
    const int gc = gCol0 + c * 16 + colL;
    const float itT = 1.0f / tauT[gc];
    const float bt  = bTn[gc];
    float gTs = 0.0f, aTs = 0.0f;
    #pragma unroll
    for (int r = 0; r < 8; ++r) {
      const float s = acc[c][r];
      const bool diag = (rowBaseW + r) == gc;
      const float eI = diag ? 0.0f : __expf(s * itI[r] - bI8[r]);
      const float eT = diag ? 0.0f : __expf(s * itT - bt);
      gIs[r] += eI; aIs[r] += eI * s;
      gTs    += eT; aTs    += eT * s;
    }
    gTs += __shfl_xor(gTs, 16, 32);
    aTs += __shfl_xor(aTs, 16, 32);
    if (lane < 16) { atomicAdd(GT + gc, gTs); atomicAdd(AT + gc, aTs); }
  }

  #pragma unroll
  for (int r = 0; r < 8; ++r) {
    float g = gIs[r], a = aIs[r];
    g += __shfl_xor(g, 8, 32); a += __shfl_xor(a, 8, 32);
    g += __shfl_xor(g, 4, 32); a += __shfl_xor(a, 4, 32);
    g += __shfl_xor(g, 2, 32); a += __shfl_xor(a, 2, 32);
    g += __shfl_xor(g, 1, 32); a += __shfl_xor(a, 1, 32);
    if ((lane & 15) == 0) {
      atomicAdd(GI + rowBaseW + r, g);
      atomicAdd(AI + rowBaseW + r, a);
    }
  }
}

__global__ void diag_kernel(const float* __restrict__ img,
                            const float* __restrict__ txt, int D,
                            float* __restrict__ diag, int B) {
  const int wave = threadIdx.x >> 5;
  const int lane = threadIdx.x & 31;
  const int row = blockIdx.x * 8 + wave;
  if (row >= B) return;
  const float* a = img + (size_t)row * D;
  const float* b = txt + (size_t)row * D;
  float s = 0.0f;
  for (int k = lane; k < D; k += 32) s += a[k] * b[k];
  s += __shfl_xor(s, 16, 32);
  s += __shfl_xor(s, 8, 32);
  s += __shfl_xor(s, 4, 32);
  s += __shfl_xor(s, 2, 32);
  s += __shfl_xor(s, 1, 32);
  if (lane == 0) diag[row] = s;
}

__global__ void final_kernel(const float* GI, const float* AI,
                             const float* GT, const float* AT,
                             const float* bIn, const float* bTn,
                             const float* oBI, const float* oBT,
                             const float* tauI, const float* tauT,
                             const float* s_I, const float* s_T,
                             const int* iid, const int* tid_,
                             const float* diag, const int* epoch_p,
                             float* out, int B) {
  const int t = threadIdx.x;
  const float neg = (float)(B - 1);
  const int epoch = epoch_p[0];
  double c0 = 0.0, c1 = 0.0, c2 = 0.0;
  for (int i = t; i < B; i += 256) {
    float gI = GI[i] / neg;
    float si = (epoch == 0) ? gI
             : (1.0f - GAM) * s_I[iid[i]] * __expf(oBI[i] - bIn[i]) + GAM * gI;
    float il  = AI[i] / (si * neg);
    float twi = logf(si) + bIn[i] + RHO_C - AI[i] / (tauI[i] * si * neg);

    float gT = GT[i] / neg;
    float st = (epoch == 0) ? gT
             : (1.0f - GAM) * s_T[tid_[i]] * __expf(oBT[i] - bTn[i]) + GAM * gT;
    float tl  = AT[i] / (st * neg);
    float twt = logf(st) + bTn[i] + RHO_C - AT[i] / (tauT[i] * st * neg);

    c0 += (double)(il + tl - 2.0f * diag[i] + twi * tauI[i] + twt * tauT[i]);
    c1 += (double)tauI[i];
    c2 += (double)tauT[i];
  }
  __shared__ double r0[256], r1[256], r2[256];
  r0[t] = c0; r1[t] = c1; r2[t] = c2;
  __syncthreads();
  for (int s = 128; s; s >>= 1) {
    if (t < s) { r0[t] += r0[t + s]; r1[t] += r1[t + s]; r2[t] += r2[t + s]; }
    __syncthreads();
  }
  if (t == 0) {
    out[0] = (float)(r0[0] / (double)B);
    out[1] = (float)(r1[0] / (double)B);
    out[2] = (float)(r2[0] / (double)B);
  }
}

extern "C" void kernel_launch(void* const* d_in, const int* in_sizes, int n_in,
                              void* d_out, int out_size, void* d_ws, size_t ws_size,
                              hipStream_t stream) {
  (void)n_in; (void)out_size; (void)ws_size;
  const float* img   = (const float*)d_in[0];
  const float* txt   = (const float*)d_in[1];
  const float* tauI  = (const float*)d_in[2];
  const float* tauT  = (const float*)d_in[3];
  const float* s_I   = (const float*)d_in[4];
  const float* s_T   = (const float*)d_in[5];
  const float* b_I   = (const float*)d_in[6];
  const float* b_T   = (const float*)d_in[7];
  const int*   iid   = (const int*)d_in[8];
  const int*   tid   = (const int*)d_in[9];
  const int*   epoch = (const int*)d_in[10];

  const int B = in_sizes[2];            // tau_image has B elements
  const int D = in_sizes[0] / B;        // image_features is [B, D]
  float* out = (float*)d_out;

  float* ws = (float*)d_ws;
  float* rowmax = ws; ws += B;
  float* colmax = ws; ws += B;
  float* bIn    = ws; ws += B;
  float* bTn    = ws; ws += B;
  float* oBI    = ws; ws += B;
  float* oBT    = ws; ws += B;
  float* GI     = ws; ws += B;
  float* AI     = ws; ws += B;
  float* GT     = ws; ws += B;
  float* AT     = ws; ws += B;
  float* diag   = ws; ws += B;

  const int nb = (B + 255) / 256;
  init_ws_kernel<<<nb, 256, 0, stream>>>(rowmax, colmax, GI, AI, GT, AT, B);

  dim3 grid(B / 64, B / 64);
  pass1_maxes<<<grid, 128, 0, stream>>>(img, txt, D, rowmax, colmax);

  finalize_b_kernel<<<nb, 256, 0, stream>>>(rowmax, colmax, tauI, tauT, b_I, b_T,
                                            iid, tid, bIn, bTn, oBI, oBT, B);

  pass2_sums<<<grid, 128, 0, stream>>>(img, txt, D, tauI, tauT, bIn, bTn,
                                       GI, AI, GT, AT);

  diag_kernel<<<(B + 7) / 8, 256, 0, stream>>>(img, txt, D, diag, B);

  final_kernel<<<1, 256, 0, stream>>>(GI, AI, GT, AT, bIn, bTn, oBI, oBT,
                                      tauI, tauT, s_I, s_T, iid, tid,
                                      diag, epoch, out, B);
}